// vivi_block_87600152969239
// MI455X (gfx1250) — compile-verified
//
#include <hip/hip_runtime.h>
#include <math.h>

#define D_MODEL 192
#define D_INNER 384
#define D_STATE 16
#define D_CONVK 4
#define DT_RANK 12
#define BATCH 2
#define SEQ 2048
#define ROWS (BATCH * SEQ)                 // 4096
#define XZ_N (2 * D_INNER)                 // 768
#define XDBL_N (DT_RANK + 2 * D_STATE)     // 44

typedef __attribute__((ext_vector_type(2))) float v2f;
typedef __attribute__((ext_vector_type(8))) float v8f;

__device__ __forceinline__ v8f wmma4(v2f a, v2f b, v8f c) {
  // D = A(16x4,f32) * B(4x16,f32) + C(16x16,f32), native f32 WMMA
  return __builtin_amdgcn_wmma_f32_16x16x4_f32(
      /*neg_a=*/false, a, /*neg_b=*/false, b,
      /*c_mod=*/(short)0, c, /*reuse_a=*/false, /*reuse_b=*/false);
}

__device__ __forceinline__ float silu_f(float x) {
  return x / (1.f + __expf(-x));
}

// ---------------------------------------------------------------------------
// 1) residual add + RMSNorm.  One wave per row (192 cols -> 6 per lane).
// ---------------------------------------------------------------------------
__global__ void k_rmsnorm(const float* __restrict__ hs, const float* __restrict__ res,
                          const float* __restrict__ nw, float* __restrict__ resid_out,
                          float* __restrict__ xnorm) {
  int row  = blockIdx.x * 4 + (threadIdx.x >> 5);
  int lane = threadIdx.x & 31;
  size_t base = (size_t)row * D_MODEL;
  float v[6];
  float ss = 0.f;
#pragma unroll
  for (int i = 0; i < 6; ++i) {
    int c = lane + 32 * i;
    float t = hs[base + c] + res[base + c];
    v[i] = t;
    resid_out[base + c] = t;
    ss += t * t;
  }
#pragma unroll
  for (int off = 16; off; off >>= 1) ss += __shfl_xor(ss, off, 32);
  float inv = rsqrtf(ss * (1.f / D_MODEL) + 1e-5f);
#pragma unroll
  for (int i = 0; i < 6; ++i) {
    int c = lane + 32 * i;
    xnorm[base + c] = v[i] * inv * nw[c];
  }
}

// ---------------------------------------------------------------------------
// 2) in_proj: (ROWS x 192) @ (768 x 192)^T -> (ROWS x 768) via WMMA f32 16x16x4
// ---------------------------------------------------------------------------
__global__ void k_inproj(const float* __restrict__ X, const float* __restrict__ W,
                         float* __restrict__ Y) {
  int gw   = (blockIdx.x * blockDim.x + threadIdx.x) >> 5;
  int lane = threadIdx.x & 31;
  const int NT = XZ_N / 16;
  int mt = gw / NT, nt = gw % NT;
  int half = lane >> 4, ln = lane & 15;
  int m = mt * 16 + ln;
  int n = nt * 16 + ln;
  v8f c = {};
  const float* arow = X + (size_t)m * D_MODEL;
  const float* brow = W + (size_t)n * D_MODEL;
  for (int k = 0; k < D_MODEL; k += 4) {
    int kk = k + 2 * half;
    v2f a = *(const v2f*)(arow + kk);
    v2f b = *(const v2f*)(brow + kk);
    c = wmma4(a, b, c);
  }
  int mbase = mt * 16 + 8 * half;
#pragma unroll
  for (int j = 0; j < 8; ++j)
    Y[(size_t)(mbase + j) * XZ_N + n] = c[j];
}

// ---------------------------------------------------------------------------
// 3) depthwise causal conv (k=4) + SiLU.  dir=0: y[t]=sum_k w[k]*x[t-3+k]
//                                         dir=1: y[t]=sum_k w[k]*x[t+3-k]
//    reads xh = xz[:, 0:384]
// ---------------------------------------------------------------------------
__global__ void k_conv(const float* __restrict__ xz, const float* __restrict__ cw,
                       const float* __restrict__ cb, float* __restrict__ out, int dir) {
  int idx = blockIdx.x * blockDim.x + threadIdx.x;
  int d = idx % D_INNER;
  int t = (idx / D_INNER) % SEQ;
  int b = idx / (D_INNER * SEQ);
  float acc = cb[d];
#pragma unroll
  for (int k = 0; k < D_CONVK; ++k) {
    int tt = (dir == 0) ? (t - (D_CONVK - 1) + k) : (t + (D_CONVK - 1) - k);
    if (tt >= 0 && tt < SEQ)
      acc += cw[d * D_CONVK + k] * xz[((size_t)(b * SEQ + tt)) * XZ_N + d];
  }
  out[idx] = silu_f(acc);
}

// ---------------------------------------------------------------------------
// 4) x_proj: (ROWS x 384) @ (44 x 384)^T -> (ROWS x 44), N padded to 48
// ---------------------------------------------------------------------------
__global__ void k_xproj(const float* __restrict__ X, const float* __restrict__ W,
                        float* __restrict__ Y) {
  int gw   = (blockIdx.x * blockDim.x + threadIdx.x) >> 5;
  int lane = threadIdx.x & 31;
  const int NT = 3;  // 48 padded cols
  int mt = gw / NT, nt = gw % NT;
  int half = lane >> 4, ln = lane & 15;
  int m = mt * 16 + ln;
  int n = nt * 16 + ln;
  bool nvalid = (n < XDBL_N);
  v8f c = {};
  const float* arow = X + (size_t)m * D_INNER;
  const float* brow = W + (size_t)n * D_INNER;
  for (int k = 0; k < D_INNER; k += 4) {
    int kk = k + 2 * half;
    v2f a = *(const v2f*)(arow + kk);
    v2f b = {0.f, 0.f};
    if (nvalid) b = *(const v2f*)(brow + kk);
    c = wmma4(a, b, c);
  }
  if (nvalid) {
    int mbase = mt * 16 + 8 * half;
#pragma unroll
    for (int j = 0; j < 8; ++j)
      Y[(size_t)(mbase + j) * XDBL_N + n] = c[j];
  }
}

// ---------------------------------------------------------------------------
// 5) dt: (ROWS x 12) @ (384 x 12)^T + bias -> softplus -> (ROWS x 384)
// ---------------------------------------------------------------------------
__global__ void k_dtproj(const float* __restrict__ XD, const float* __restrict__ W,
                         const float* __restrict__ bias, float* __restrict__ Y) {
  int gw   = (blockIdx.x * blockDim.x + threadIdx.x) >> 5;
  int lane = threadIdx.x & 31;
  const int NT = D_INNER / 16;  // 24
  int mt = gw / NT, nt = gw % NT;
  int half = lane >> 4, ln = lane & 15;
  int m = mt * 16 + ln;
  int n = nt * 16 + ln;
  v8f c = {};
  const float* arow = XD + (size_t)m * XDBL_N;  // cols 0..11 are dt-rank slice
  const float* brow = W + (size_t)n * DT_RANK;
  for (int k = 0; k < DT_RANK; k += 4) {
    int kk = k + 2 * half;
    v2f a = *(const v2f*)(arow + kk);
    v2f b = *(const v2f*)(brow + kk);
    c = wmma4(a, b, c);
  }
  int mbase = mt * 16 + 8 * half;
  float bv = bias[n];
#pragma unroll
  for (int j = 0; j < 8; ++j) {
    float v = c[j] + bv;
    // stable softplus
    v = fmaxf(v, 0.f) + log1pf(__expf(-fabsf(v)));
    Y[(size_t)(mbase + j) * D_INNER + n] = v;
  }
}

// ---------------------------------------------------------------------------
// 6) selective scan. Lane = (channel-half, state n). Each 16-lane group owns
//    one channel d of one batch; state h kept in a register, 16-lane shfl_xor
//    tree contracts h*C each step. dir selects time order; output already
//    placed at the true time index and gated with SiLU(z).
// ---------------------------------------------------------------------------
__global__ void k_scan(const float* __restrict__ convx, const float* __restrict__ xdbl,
                       const float* __restrict__ dt, const float* __restrict__ xz,
                       const float* __restrict__ A_log, const float* __restrict__ Dp,
                       float* __restrict__ ybuf, int dir) {
  int lane = threadIdx.x & 31;
  int waveInBlock = threadIdx.x >> 5;
  int chan = blockIdx.x * 16 + waveInBlock * 2 + (lane >> 4);  // 0 .. 2*384-1
  int b = chan / D_INNER;
  int d = chan % D_INNER;
  int n = lane & 15;
  float Acoef = -__expf(A_log[d * D_STATE + n]);
  float Dd = Dp[d];
  float h = 0.f;
  size_t base_b = (size_t)b * SEQ;
  for (int t = 0; t < SEQ; ++t) {
    int tt = dir ? (SEQ - 1 - t) : t;
    size_t row = base_b + tt;
    float dtv = dt[row * D_INNER + d];
    float xv  = convx[row * D_INNER + d];
    float Bv  = xdbl[row * XDBL_N + DT_RANK + n];
    float Cv  = xdbl[row * XDBL_N + DT_RANK + D_STATE + n];
    h = __expf(dtv * Acoef) * h + (dtv * xv) * Bv;
    float acc = h * Cv;
    acc += __shfl_xor(acc, 8, 32);
    acc += __shfl_xor(acc, 4, 32);
    acc += __shfl_xor(acc, 2, 32);
    acc += __shfl_xor(acc, 1, 32);
    if (n == 0) {
      float zv = xz[row * XZ_N + D_INNER + d];
      float y = (acc + xv * Dd) * silu_f(zv);
      ybuf[row * D_INNER + d] = y;
    }
  }
}

// ---------------------------------------------------------------------------
// 7) out_proj: (y_f + y_b)(ROWS x 384) @ (192 x 384)^T -> (ROWS x 192)
// ---------------------------------------------------------------------------
__global__ void k_outproj(const float* __restrict__ YF, const float* __restrict__ YB,
                          const float* __restrict__ W, float* __restrict__ O) {
  int gw   = (blockIdx.x * blockDim.x + threadIdx.x) >> 5;
  int lane = threadIdx.x & 31;
  const int NT = D_MODEL / 16;  // 12
  int mt = gw / NT, nt = gw % NT;
  int half = lane >> 4, ln = lane & 15;
  int m = mt * 16 + ln;
  int n = nt * 16 + ln;
  v8f c = {};
  const float* frow = YF + (size_t)m * D_INNER;
  const float* grow = YB + (size_t)m * D_INNER;
  const float* brow = W + (size_t)n * D_INNER;
  for (int k = 0; k < D_INNER; k += 4) {
    int kk = k + 2 * half;
    v2f f = *(const v2f*)(frow + kk);
    v2f g = *(const v2f*)(grow + kk);
    v2f a;
    a.x = f.x + g.x;
    a.y = f.y + g.y;
    v2f b = *(const v2f*)(brow + kk);
    c = wmma4(a, b, c);
  }
  int mbase = mt * 16 + 8 * half;
#pragma unroll
  for (int j = 0; j < 8; ++j)
    O[(size_t)(mbase + j) * D_MODEL + n] = c[j];
}

// ---------------------------------------------------------------------------
extern "C" void kernel_launch(void* const* d_in, const int* in_sizes, int n_in,
                              void* d_out, int out_size, void* d_ws, size_t ws_size,
                              hipStream_t stream) {
  const float* hs   = (const float*)d_in[0];
  const float* res  = (const float*)d_in[1];
  const float* nw   = (const float*)d_in[2];
  const float* inW  = (const float*)d_in[3];
  const float* outW = (const float*)d_in[4];
  const float* cwf  = (const float*)d_in[5];
  const float* cbf  = (const float*)d_in[6];
  const float* xpf  = (const float*)d_in[7];
  const float* dwf  = (const float*)d_in[8];
  const float* dbf  = (const float*)d_in[9];
  const float* Alf  = (const float*)d_in[10];
  const float* Df   = (const float*)d_in[11];
  const float* cwb  = (const float*)d_in[12];
  const float* cbb  = (const float*)d_in[13];
  const float* xpb  = (const float*)d_in[14];
  const float* dwb  = (const float*)d_in[15];
  const float* dbb  = (const float*)d_in[16];
  const float* Alb  = (const float*)d_in[17];
  const float* Db   = (const float*)d_in[18];

  float* out       = (float*)d_out;                       // (out, residual_out) concatenated
  float* resid_out = out + (size_t)ROWS * D_MODEL;

  float* ws    = (float*)d_ws;
  float* xnorm = ws;                                      // ROWS*192
  float* xz    = xnorm + (size_t)ROWS * D_MODEL;          // ROWS*768
  float* conv  = xz + (size_t)ROWS * XZ_N;                // ROWS*384 (reused per dir)
  float* xdbl  = conv + (size_t)ROWS * D_INNER;           // ROWS*44  (reused per dir)
  float* dtb   = xdbl + (size_t)ROWS * XDBL_N;            // ROWS*384 (reused per dir)
  float* yf    = dtb + (size_t)ROWS * D_INNER;            // ROWS*384
  float* yb    = yf + (size_t)ROWS * D_INNER;             // ROWS*384

  // 1) residual add + RMSNorm
  k_rmsnorm<<<ROWS / 4, 128, 0, stream>>>(hs, res, nw, resid_out, xnorm);
  // 2) in_proj GEMM (WMMA)
  k_inproj<<<(ROWS / 16) * (XZ_N / 16) / 8, 256, 0, stream>>>(xnorm, inW, xz);

  // forward direction
  k_conv<<<(ROWS * D_INNER) / 256, 256, 0, stream>>>(xz, cwf, cbf, conv, 0);
  k_xproj<<<(ROWS / 16) * 3 / 8, 256, 0, stream>>>(conv, xpf, xdbl);
  k_dtproj<<<(ROWS / 16) * (D_INNER / 16) / 8, 256, 0, stream>>>(xdbl, dwf, dbf, dtb);
  k_scan<<<(BATCH * D_INNER) / 16, 256, 0, stream>>>(conv, xdbl, dtb, xz, Alf, Df, yf, 0);

  // backward direction (reuses conv/xdbl/dt scratch; stream order guarantees safety)
  k_conv<<<(ROWS * D_INNER) / 256, 256, 0, stream>>>(xz, cwb, cbb, conv, 1);
  k_xproj<<<(ROWS / 16) * 3 / 8, 256, 0, stream>>>(conv, xpb, xdbl);
  k_dtproj<<<(ROWS / 16) * (D_INNER / 16) / 8, 256, 0, stream>>>(xdbl, dwb, dbb, dtb);
  k_scan<<<(BATCH * D_INNER) / 16, 256, 0, stream>>>(conv, xdbl, dtb, xz, Alb, Db, yb, 1);

  // 7) out_proj GEMM (WMMA), fusing y_f + y_b in the A-operand fetch
  k_outproj<<<(ROWS / 16) * (D_MODEL / 16) / 8, 256, 0, stream>>>(yf, yb, outW, out);
}